// AttentionSelector_74174085202134
// MI455X (gfx1250) — compile-verified
//
#include <hip/hip_runtime.h>

typedef __attribute__((ext_vector_type(2))) float v2f;
typedef __attribute__((ext_vector_type(8))) float v8f;

#define WMMA_F32_K4(a, b, c) \
  __builtin_amdgcn_wmma_f32_16x16x4_f32(false, (a), false, (b), (short)0, (c), false, false)

#define D_MODEL 512
#define D_K     64
#define NKEYS   32768
#define NB      4096
#define NACT    64
#define EPS     1e-8f

// ---------------------------------------------------------------------------
// out[M, 64] = X[M, 512] @ W[64, 512]^T + bias
// grid.x = M/16 ; block = 128 (4 waves); wave w computes the 16x16 tile at n0 = 16*w
// ---------------------------------------------------------------------------
__global__ __launch_bounds__(128) void proj_kernel(const float* __restrict__ X,
                                                   const float* __restrict__ W,
                                                   const float* __restrict__ bias,
                                                   float* __restrict__ out) {
  __shared__ float xs[16][D_MODEL + 4];
  const int tid = threadIdx.x;
  const int m0  = blockIdx.x * 16;

  // Stage the 16x512 X tile (2048 float4 = 128 threads * 16)
  const float* src = X + (size_t)m0 * D_MODEL;
  for (int i = tid; i < 16 * (D_MODEL / 4); i += 128) {
    const int r = i / (D_MODEL / 4);
    const int c = i % (D_MODEL / 4);
    const float4 v = ((const float4*)(src + (size_t)r * D_MODEL))[c];
    xs[r][c * 4 + 0] = v.x; xs[r][c * 4 + 1] = v.y;
    xs[r][c * 4 + 2] = v.z; xs[r][c * 4 + 3] = v.w;
  }
  __syncthreads();

  const int wave = tid >> 5, lane = tid & 31;
  const int n0   = wave * 16;
  const int rc   = lane & 15;        // A: M index / B: N index
  const int koff = (lane >> 4) * 2;  // K pair selector

  const float* wrow = W + (size_t)(n0 + rc) * D_MODEL;  // B[k,n] = W[n,k]
  v8f acc = {};
  for (int k0 = 0; k0 < D_MODEL; k0 += 4) {
    v2f a, b;
    a.x = xs[rc][k0 + koff];
    a.y = xs[rc][k0 + koff + 1];
    b.x = wrow[k0 + koff];
    b.y = wrow[k0 + koff + 1];
    acc = WMMA_F32_K4(a, b, acc);
  }

  const int mh = 8 * (lane >> 4);
  const int n  = n0 + rc;
  const float bv = bias[n];
#pragma unroll
  for (int i = 0; i < 8; ++i)
    out[(size_t)(m0 + mh + i) * D_K + n] = acc[i] + bv;
}

// ---------------------------------------------------------------------------
// v[n][a] = Wv_w[a][actions[n]] + Wv_b[a]   (one_hot @ Wv^T == row gather)
// ---------------------------------------------------------------------------
__global__ void gather_v_kernel(const int* __restrict__ actions,
                                const float* __restrict__ Wv_w,
                                const float* __restrict__ Wv_b,
                                float* __restrict__ v) {
  const int idx = blockIdx.x * blockDim.x + threadIdx.x;
  if (idx >= NKEYS * NACT) return;
  const int n = idx >> 6, a = idx & 63;
  const int act = actions[n];
  v[idx] = Wv_w[a * NACT + act] + Wv_b[a];
}

// ---------------------------------------------------------------------------
// attn[m, n] = (qp[m,:] . kp[n,:]) / 8    -> written into d_out sparse region
// block = 256 (8 waves); tile 16 rows x 128 cols; grid = (N/128, B/16)
// ---------------------------------------------------------------------------
__global__ __launch_bounds__(256) void attn_kernel(const float* __restrict__ qp,
                                                   const float* __restrict__ kp,
                                                   float* __restrict__ attn_out) {
  __shared__ float qs[16][D_K + 4];
  const int tid = threadIdx.x;
  const int m0  = blockIdx.y * 16;

  for (int i = tid; i < 16 * (D_K / 4); i += 256) {
    const int r = i / (D_K / 4);
    const int c = i % (D_K / 4);
    const float4 v = ((const float4*)(qp + (size_t)(m0 + r) * D_K))[c];
    qs[r][c * 4 + 0] = v.x; qs[r][c * 4 + 1] = v.y;
    qs[r][c * 4 + 2] = v.z; qs[r][c * 4 + 3] = v.w;
  }
  __syncthreads();

  const int wave = tid >> 5, lane = tid & 31;
  const int n0   = blockIdx.x * 128 + wave * 16;
  const int rc   = lane & 15;
  const int koff = (lane >> 4) * 2;

  const float* krow = kp + (size_t)(n0 + rc) * D_K;  // B[k,n] = kp[n,k]
  v8f acc = {};
#pragma unroll
  for (int k0 = 0; k0 < D_K; k0 += 4) {
    v2f a, b;
    a.x = qs[rc][k0 + koff];
    a.y = qs[rc][k0 + koff + 1];
    b.x = krow[k0 + koff];
    b.y = krow[k0 + koff + 1];
    acc = WMMA_F32_K4(a, b, acc);
  }

  const float inv_temp = 0.125f;  // 1/sqrt(64)
  const int mh = 8 * (lane >> 4);
  const size_t base = (size_t)(m0 + mh) * NKEYS + (size_t)(n0 + rc);
#pragma unroll
  for (int i = 0; i < 8; ++i)
    attn_out[base + (size_t)i * NKEYS] = acc[i] * inv_temp;
}

// ---------------------------------------------------------------------------
// Per-row top-16 threshold + renormalize (in place) + final_attn = sparse @ v
// One block (256 threads) per row. Register-resident sorted top-16 per thread,
// 8 LDS merge levels, then one patch pass over the <=16 survivors.
// ---------------------------------------------------------------------------
#define TK   16
#define TPB  256
#define CAP  256

__global__ __launch_bounds__(TPB) void topk_kernel(float* __restrict__ attn,
                                                   const float* __restrict__ v,
                                                   float* __restrict__ final_out) {
  __shared__ float s_lists[TPB * TK];
  __shared__ int   s_idx[CAP];
  __shared__ float s_w[CAP];
  __shared__ float s_sum;
  __shared__ float s_delta;
  __shared__ int   s_count;

  const int tid = threadIdx.x;
  const int row = blockIdx.x;
  float* arow = attn + (size_t)row * NKEYS;

  if (tid == 0) { s_sum = 0.0f; s_count = 0; }

  // ---- scan 1: per-thread sorted-descending top-16 of 128 strided elements
  float t[TK];
#pragma unroll
  for (int i = 0; i < TK; ++i) t[i] = -3.4e38f;

  for (int i = tid; i < NKEYS; i += TPB) {
    const float x = arow[i];
    if (x > t[TK - 1]) {
      t[TK - 1] = x;
#pragma unroll
      for (int j = TK - 1; j > 0; --j) {
        if (t[j] > t[j - 1]) { const float tmp = t[j]; t[j] = t[j - 1]; t[j - 1] = tmp; }
      }
    }
  }
#pragma unroll
  for (int i = 0; i < TK; ++i) s_lists[tid * TK + i] = t[i];
  __syncthreads();

  // ---- log-merge 256 lists down to one (insert-merge keeps static reg idx)
  for (int s = TPB / 2; s >= 1; s >>= 1) {
    if (tid < s) {
      const float* other = &s_lists[(tid + s) * TK];
      for (int j = 0; j < TK; ++j) {
        const float x = other[j];
        if (x <= t[TK - 1]) break;  // sorted descending: rest are smaller
        t[TK - 1] = x;
#pragma unroll
        for (int k = TK - 1; k > 0; --k) {
          if (t[k] > t[k - 1]) { const float tmp = t[k]; t[k] = t[k - 1]; t[k - 1] = tmp; }
        }
      }
#pragma unroll
      for (int i = 0; i < TK; ++i) s_lists[tid * TK + i] = t[i];
    }
    __syncthreads();
  }
  if (tid == 0) s_delta = s_lists[TK - 1] + EPS;  // 16th largest + eps
  __syncthreads();

  // ---- scan 2: write raw clipped weights, collect survivors, reduce sum
  const float delta = s_delta;
  float part = 0.0f;
  for (int i = tid; i < NKEYS; i += TPB) {
    const float w = fmaxf(arow[i] - delta, 0.0f);
    arow[i] = w;  // survivors get patched with normalized value below
    if (w > 0.0f) {
      part += w;
      const int p = atomicAdd(&s_count, 1);
      if (p < CAP) { s_idx[p] = i; s_w[p] = w; }
    }
  }
  if (part != 0.0f) atomicAdd(&s_sum, part);
  __syncthreads();

  const float inv = 1.0f / (s_sum + EPS);
  const int cnt = (s_count < CAP) ? s_count : CAP;

  // patch survivors in place with normalized weight
  if (tid < cnt) arow[s_idx[tid]] = s_w[tid] * inv;

  // final_attn[row, a] = sum_j wnorm_j * v[idx_j, a]
  if (tid < NACT) {
    float acc = 0.0f;
    for (int j = 0; j < cnt; ++j)
      acc += s_w[j] * v[(size_t)s_idx[j] * NACT + tid];
    final_out[(size_t)row * NACT + tid] = acc * inv;
  }
}

// ---------------------------------------------------------------------------
extern "C" void kernel_launch(void* const* d_in, const int* in_sizes, int n_in,
                              void* d_out, int out_size, void* d_ws, size_t ws_size,
                              hipStream_t stream) {
  const float* q        = (const float*)d_in[0];
  const float* k_enc    = (const float*)d_in[1];
  const int*   k_act    = (const int*)d_in[2];
  const float* Wq_w     = (const float*)d_in[3];
  const float* Wq_b     = (const float*)d_in[4];
  const float* Wk_w     = (const float*)d_in[5];
  const float* Wk_b     = (const float*)d_in[6];
  const float* Wv_w     = (const float*)d_in[7];
  const float* Wv_b     = (const float*)d_in[8];

  float* final_out = (float*)d_out;                 // [4096, 64]
  float* sparse    = final_out + (size_t)NB * NACT; // [4096, 32768]

  float* ws = (float*)d_ws;
  float* qp = ws;                                   // [4096, 64]
  float* kp = qp + (size_t)NB * D_K;                // [32768, 64]
  float* vv = kp + (size_t)NKEYS * D_K;             // [32768, 64]

  proj_kernel<<<dim3(NB / 16), 128, 0, stream>>>(q, Wq_w, Wq_b, qp);
  proj_kernel<<<dim3(NKEYS / 16), 128, 0, stream>>>(k_enc, Wk_w, Wk_b, kp);
  gather_v_kernel<<<dim3((NKEYS * NACT) / 256), 256, 0, stream>>>(k_act, Wv_w, Wv_b, vv);
  attn_kernel<<<dim3(NKEYS / 128, NB / 16), 256, 0, stream>>>(qp, kp, sparse);
  topk_kernel<<<dim3(NB), TPB, 0, stream>>>(sparse, vv, final_out);
}